// DropGCN_79370995630764
// MI455X (gfx1250) — compile-verified
//
#include <hip/hip_runtime.h>
#include <hip/hip_bf16.h>
#include <math.h>

#define P_   4
#define N_   50000
#define E_   800000
#define H_   64
#define OUT_ 10
#define G_   512
#define NT   (P_ * N_)      // 200000 rows
#define EPS_ 1e-5f

typedef __attribute__((ext_vector_type(2))) float v2f;
typedef __attribute__((ext_vector_type(8))) float v8f;

// ---------------- utility kernels ----------------
__global__ void zero_f32_kernel(float* p, int n) {
    int i = blockIdx.x * blockDim.x + threadIdx.x;
    if (i < n) p[i] = 0.0f;
}

__global__ void zero_i32_kernel(int* p, int n) {
    int i = blockIdx.x * blockDim.x + threadIdx.x;
    if (i < n) p[i] = 0;
}

// max over 2*E edge indices -> mx[0]
__global__ void edge_max_kernel(const int* __restrict__ ei, int n, int* mx) {
    int i = blockIdx.x * blockDim.x + threadIdx.x;
    int v = (i < n) ? ei[i] : 0;
    #pragma unroll
    for (int off = 16; off > 0; off >>= 1)
        v = max(v, __shfl_down(v, off));
    if ((threadIdx.x & 31) == 0) atomicMax(mx, v);
}

__global__ void finalize_offset_kernel(const int* mx, int* offp) {
    if (threadIdx.x == 0 && blockIdx.x == 0) offp[0] = mx[0] + 1;
}

// ---------------- graph normalization ----------------
__global__ void degree_kernel(const int* __restrict__ ei, const int* __restrict__ offp,
                              float* __restrict__ deg) {
    long t = (long)blockIdx.x * blockDim.x + threadIdx.x;
    if (t >= (long)P_ * E_) return;
    int p = (int)(t / E_);
    int e = (int)(t % E_);
    int off = offp[0];
    int dst = p * off + ei[E_ + e];
    atomicAdd(&deg[dst], 1.0f);
}

__global__ void dinv_kernel(float* deg) {  // in place: deg -> rsqrt(deg+1)
    int i = blockIdx.x * blockDim.x + threadIdx.x;
    if (i < NT) deg[i] = rsqrtf(deg[i] + 1.0f);
}

// ---------------- node-drop init + layer-0 pooling ----------------
__global__ void init_h_kernel(const float* __restrict__ x, const int* __restrict__ mask,
                              const int* __restrict__ batch, float* __restrict__ h,
                              float* __restrict__ pooled0) {
    long t = (long)blockIdx.x * blockDim.x + threadIdx.x;
    if (t >= (long)NT * 16) return;
    long i = t >> 4;
    int  q = (int)(t & 15);
    int  n = (int)(i % N_);
    float4 v;
    if (mask[i]) { v.x = v.y = v.z = v.w = 0.0f; }
    else         { v = *(const float4*)&x[(long)n * H_ + q * 4]; }
    *(float4*)&h[i * H_ + q * 4] = v;
    int g = batch[n];
    float* pp = &pooled0[(long)g * H_ + q * 4];
    atomicAdd(pp + 0, 0.25f * v.x);
    atomicAdd(pp + 1, 0.25f * v.y);
    atomicAdd(pp + 2, 0.25f * v.z);
    atomicAdd(pp + 3, 0.25f * v.w);
}

// ---------------- fp32 WMMA GEMM: HW[NT,64] = H[NT,64] @ W[64,64] ----------------
// 256 threads = 8 waves; each wave owns TWO 16-row tiles (32 rows) and all four
// 16-col tiles, so B fragments are reused across both A tiles.
// LDS W layout is pair-interleaved: Wl[(k>>1)*128 + n*2 + (k&1)] so one B
// fragment (W[kb][n], W[kb+1][n]) is a single 8-byte ds_load_b64 into an
// even-aligned VGPR pair (no repacking movs).
// A frag (16x4 f32, 2 VGPR): lanes 0-15 hold K={k0,k0+1}, lanes 16-31 K={k0+2,k0+3}.
// D (16x16 f32, 8 VGPR): VGPR r -> M = r + (lane>=16 ? 8 : 0), N = lane&15.
__global__ __launch_bounds__(256) void gemm_h_w_kernel(const float* __restrict__ Hh,
                                                       const float* __restrict__ W,
                                                       float* __restrict__ HW) {
    __shared__ float Wl[H_ * H_];
    int tid = threadIdx.x;
    for (int idx = tid; idx < H_ * H_; idx += 256) {
        int k = idx >> 6;
        int n = idx & 63;
        Wl[(k >> 1) * 128 + n * 2 + (k & 1)] = W[idx];
    }
    __syncthreads();

    int wave = tid >> 5;
    int lane = tid & 31;
    long rb0 = ((long)blockIdx.x * 8 + wave) * 32;
    if (rb0 >= NT) return;
    long rb1 = rb0 + 16;
    bool has1 = (rb1 < NT);          // NT is a multiple of 16

    int m    = lane & 15;
    int koff = (lane >> 4) << 1;     // 0 for lanes 0-15, 2 for lanes 16-31
    const float* arow0 = &Hh[(rb0 + m) * H_ + koff];
    const float* arow1 = &Hh[((has1 ? rb1 : rb0) + m) * H_ + koff];

    v8f a00 = {}, a01 = {}, a02 = {}, a03 = {};
    v8f a10 = {}, a11 = {}, a12 = {}, a13 = {};

    #pragma unroll
    for (int k0 = 0; k0 < H_; k0 += 4) {
        v2f af0 = *(const v2f*)&arow0[k0];
        v2f af1 = *(const v2f*)&arow1[k0];
        int kp = (k0 + koff) >> 1;
        const v2f* wp = (const v2f*)&Wl[kp * 128];
        v2f b0 = wp[m];
        v2f b1 = wp[m + 16];
        v2f b2 = wp[m + 32];
        v2f b3 = wp[m + 48];
        a00 = __builtin_amdgcn_wmma_f32_16x16x4_f32(false, af0, false, b0, (short)0, a00, false, false);
        a01 = __builtin_amdgcn_wmma_f32_16x16x4_f32(false, af0, false, b1, (short)0, a01, false, false);
        a02 = __builtin_amdgcn_wmma_f32_16x16x4_f32(false, af0, false, b2, (short)0, a02, false, false);
        a03 = __builtin_amdgcn_wmma_f32_16x16x4_f32(false, af0, false, b3, (short)0, a03, false, false);
        a10 = __builtin_amdgcn_wmma_f32_16x16x4_f32(false, af1, false, b0, (short)0, a10, false, false);
        a11 = __builtin_amdgcn_wmma_f32_16x16x4_f32(false, af1, false, b1, (short)0, a11, false, false);
        a12 = __builtin_amdgcn_wmma_f32_16x16x4_f32(false, af1, false, b2, (short)0, a12, false, false);
        a13 = __builtin_amdgcn_wmma_f32_16x16x4_f32(false, af1, false, b3, (short)0, a13, false, false);
    }

    int rlo = (lane >> 4) << 3;      // M offset 0 / 8 per lane half
    #pragma unroll
    for (int r = 0; r < 8; ++r) {
        float* d0 = &HW[(rb0 + rlo + r) * H_ + m];
        d0[0]  = a00[r];
        d0[16] = a01[r];
        d0[32] = a02[r];
        d0[48] = a03[r];
    }
    if (has1) {
        #pragma unroll
        for (int r = 0; r < 8; ++r) {
            float* d1 = &HW[(rb1 + rlo + r) * H_ + m];
            d1[0]  = a10[r];
            d1[16] = a11[r];
            d1[32] = a12[r];
            d1[48] = a13[r];
        }
    }
}

// ---------------- edge scatter-add: AGG[dst] += dinv[s]*dinv[d] * HW[src] ----------------
__global__ void aggregate_kernel(const int* __restrict__ ei, const int* __restrict__ offp,
                                 const float* __restrict__ dinv, const float* __restrict__ HW,
                                 float* __restrict__ AGG) {
    long t = (long)blockIdx.x * blockDim.x + threadIdx.x;
    if (t >= (long)P_ * E_ * 16) return;
    int  q  = (int)(t & 15);
    long eg = t >> 4;
    int  e  = (int)(eg % E_);
    int  p  = (int)(eg / E_);
    int off = offp[0];
    int s = p * off + ei[e];
    int d = p * off + ei[E_ + e];
    float w = dinv[s] * dinv[d];
    float4 v = *(const float4*)&HW[(long)s * H_ + q * 4];
    float* o = &AGG[(long)d * H_ + q * 4];
    atomicAdd(o + 0, w * v.x);
    atomicAdd(o + 1, w * v.y);
    atomicAdd(o + 2, w * v.z);
    atomicAdd(o + 3, w * v.w);
}

// ---------------- self-loop + bias (in place on AGG) ----------------
__global__ void combine_kernel(const float* __restrict__ dinv, const float* __restrict__ HW,
                               const float* __restrict__ bias, float* __restrict__ AGG) {
    long t = (long)blockIdx.x * blockDim.x + threadIdx.x;
    if (t >= (long)NT * 16) return;
    long i = t >> 4;
    int  q = (int)(t & 15);
    float ns = dinv[i] * dinv[i];
    float4 hw = *(const float4*)&HW[i * H_ + q * 4];
    float4 a  = *(const float4*)&AGG[i * H_ + q * 4];
    float4 b  = *(const float4*)&bias[q * 4];
    a.x += ns * hw.x + b.x;
    a.y += ns * hw.y + b.y;
    a.z += ns * hw.z + b.z;
    a.w += ns * hw.w + b.w;
    *(float4*)&AGG[i * H_ + q * 4] = a;
}

// ---------------- BatchNorm column stats: sums[0:64]=sum, sums[64:128]=sumsq ----------------
__global__ __launch_bounds__(256) void bn_stats_kernel(const float* __restrict__ Hn,
                                                       float* __restrict__ sums) {
    __shared__ float ssum[256], ssq[256];
    int c   = threadIdx.x & 63;
    int sub = threadIdx.x >> 6;  // 0..3
    long rpb = (NT + gridDim.x - 1) / gridDim.x;
    long r0 = (long)blockIdx.x * rpb;
    long r1 = r0 + rpb; if (r1 > NT) r1 = NT;
    float s = 0.0f, sq = 0.0f;
    for (long r = r0 + sub; r < r1; r += 4) {
        float v = Hn[r * H_ + c];
        s += v; sq += v * v;
    }
    ssum[threadIdx.x] = s;
    ssq[threadIdx.x]  = sq;
    __syncthreads();
    if (sub == 0) {
        s  = ssum[c] + ssum[c + 64] + ssum[c + 128] + ssum[c + 192];
        sq = ssq[c]  + ssq[c + 64]  + ssq[c + 128]  + ssq[c + 192];
        atomicAdd(&sums[c], s);
        atomicAdd(&sums[64 + c], sq);
    }
}

__global__ void bn_finalize_kernel(const float* __restrict__ sums, const float* __restrict__ gamma,
                                   const float* __restrict__ beta, float* __restrict__ bnp) {
    int c = threadIdx.x;
    if (c >= H_) return;
    const float inv_n = 1.0f / (float)NT;
    float mu  = sums[c] * inv_n;
    float var = sums[64 + c] * inv_n - mu * mu;
    float sc  = gamma[c] * rsqrtf(var + EPS_);
    bnp[c]      = sc;
    bnp[64 + c] = beta[c] - mu * sc;
}

// ---------------- BN + ReLU -> h, with per-layer graph pooling ----------------
__global__ void bn_relu_pool_kernel(const float* __restrict__ Hn, const float* __restrict__ bnp,
                                    const int* __restrict__ batch, float* __restrict__ Ho,
                                    float* __restrict__ pooled) {
    long t = (long)blockIdx.x * blockDim.x + threadIdx.x;
    if (t >= (long)NT * 16) return;
    long i = t >> 4;
    int  q = (int)(t & 15);
    float4 v  = *(const float4*)&Hn[i * H_ + q * 4];
    float4 sc = *(const float4*)&bnp[q * 4];
    float4 sh = *(const float4*)&bnp[64 + q * 4];
    v.x = fmaxf(0.0f, v.x * sc.x + sh.x);
    v.y = fmaxf(0.0f, v.y * sc.y + sh.y);
    v.z = fmaxf(0.0f, v.z * sc.z + sh.z);
    v.w = fmaxf(0.0f, v.w * sc.w + sh.w);
    *(float4*)&Ho[i * H_ + q * 4] = v;
    int n = (int)(i % N_);
    int g = batch[n];
    float* pp = &pooled[(long)g * H_ + q * 4];
    atomicAdd(pp + 0, 0.25f * v.x);
    atomicAdd(pp + 1, 0.25f * v.y);
    atomicAdd(pp + 2, 0.25f * v.z);
    atomicAdd(pp + 3, 0.25f * v.w);
}

// ---------------- 5-head FC heads + sum ----------------
__global__ void fc_kernel(const float* __restrict__ pooled,  // [5][G][64]
                          const float* __restrict__ fcw,     // [5][64][10]
                          const float* __restrict__ fcb,     // [5][10]
                          float* __restrict__ logits) {      // [G][10]
    int t = blockIdx.x * blockDim.x + threadIdx.x;
    if (t >= G_ * OUT_) return;
    int g = t / OUT_;
    int c = t % OUT_;
    float acc = 0.0f;
    #pragma unroll
    for (int i = 0; i < 5; ++i) {
        acc += fcb[i * OUT_ + c];
        const float* pg   = &pooled[((long)i * G_ + g) * H_];
        const float* wcol = &fcw[(long)i * H_ * OUT_ + c];
        #pragma unroll
        for (int k = 0; k < H_; ++k) acc += pg[k] * wcol[k * OUT_];
    }
    logits[t] = acc;
}

__global__ void log_softmax_kernel(const float* __restrict__ logits, float* __restrict__ out) {
    int g = blockIdx.x * blockDim.x + threadIdx.x;
    if (g >= G_) return;
    float m = -1e30f;
    #pragma unroll
    for (int c = 0; c < OUT_; ++c) m = fmaxf(m, logits[g * OUT_ + c]);
    float s = 0.0f;
    #pragma unroll
    for (int c = 0; c < OUT_; ++c) s += __expf(logits[g * OUT_ + c] - m);
    float lse = __logf(s);
    #pragma unroll
    for (int c = 0; c < OUT_; ++c) out[g * OUT_ + c] = logits[g * OUT_ + c] - m - lse;
}

// ---------------- host-side orchestration ----------------
static inline int cdiv(long a, int b) { return (int)((a + b - 1) / b); }

extern "C" void kernel_launch(void* const* d_in, const int* in_sizes, int n_in,
                              void* d_out, int out_size, void* d_ws, size_t ws_size,
                              hipStream_t stream) {
    const float* x        = (const float*)d_in[0];   // [N,64]
    const int*   ei       = (const int*)  d_in[1];   // [2,E]
    const int*   batch    = (const int*)  d_in[2];   // [N]
    const int*   mask     = (const int*)  d_in[3];   // [P,N]
    const float* conv_w   = (const float*)d_in[4];   // [4,64,64]
    const float* conv_b   = (const float*)d_in[5];   // [4,64]
    const float* bn_gamma = (const float*)d_in[6];   // [4,64]
    const float* bn_beta  = (const float*)d_in[7];   // [4,64]
    const float* fc_w     = (const float*)d_in[8];   // [5,64,10]
    const float* fc_b     = (const float*)d_in[9];   // [5,10]
    float* out = (float*)d_out;                      // [G,10]

    float* h      = (float*)d_ws;                    // NT*64
    float* hw     = h   + (long)NT * H_;             // NT*64
    float* agg    = hw  + (long)NT * H_;             // NT*64
    float* deg    = agg + (long)NT * H_;             // NT (becomes dinv)
    float* pooled = deg + NT;                        // 5*G*64
    float* bnsums = pooled + 5L * G_ * H_;           // 128
    float* bnp    = bnsums + 128;                    // 128
    float* logits = bnp + 128;                       // G*10
    int*   ipair  = (int*)(logits + G_ * OUT_);      // [0]=max, [1]=offset

    const int B = 256;

    // offset = max(edge_index) + 1
    zero_i32_kernel<<<1, 32, 0, stream>>>(ipair, 2);
    edge_max_kernel<<<cdiv(2L * E_, B), B, 0, stream>>>(ei, 2 * E_, ipair);
    finalize_offset_kernel<<<1, 1, 0, stream>>>(ipair, ipair + 1);

    // degrees -> dinv
    zero_f32_kernel<<<cdiv(NT, B), B, 0, stream>>>(deg, NT);
    degree_kernel<<<cdiv((long)P_ * E_, B), B, 0, stream>>>(ei, ipair + 1, deg);
    dinv_kernel<<<cdiv(NT, B), B, 0, stream>>>(deg);

    // pooled accumulators
    zero_f32_kernel<<<cdiv(5L * G_ * H_, B), B, 0, stream>>>(pooled, 5 * G_ * H_);

    // x_rep (node drop) + layer-0 pooling
    init_h_kernel<<<cdiv((long)NT * 16, B), B, 0, stream>>>(x, mask, batch, h, pooled);

    for (int layer = 0; layer < 4; ++layer) {
        // hw = h @ W  (fp32 WMMA, 256 rows per block)
        gemm_h_w_kernel<<<cdiv((long)NT, 256), B, 0, stream>>>(h, conv_w + (long)layer * H_ * H_, hw);
        // agg = scatter_add(norm_e * hw[src] -> dst)
        zero_f32_kernel<<<cdiv((long)NT * H_, B), B, 0, stream>>>(agg, NT * H_);
        aggregate_kernel<<<cdiv((long)P_ * E_ * 16, B), B, 0, stream>>>(ei, ipair + 1, deg, hw, agg);
        // agg += norm_s * hw + bias
        combine_kernel<<<cdiv((long)NT * 16, B), B, 0, stream>>>(deg, hw, conv_b + (long)layer * H_, agg);
        // BatchNorm (batch stats) + ReLU -> h, plus pooling into head (layer+1)
        zero_f32_kernel<<<1, 128, 0, stream>>>(bnsums, 128);
        bn_stats_kernel<<<512, B, 0, stream>>>(agg, bnsums);
        bn_finalize_kernel<<<1, 64, 0, stream>>>(bnsums, bn_gamma + (long)layer * H_,
                                                 bn_beta + (long)layer * H_, bnp);
        bn_relu_pool_kernel<<<cdiv((long)NT * 16, B), B, 0, stream>>>(
            agg, bnp, batch, h, pooled + (long)(layer + 1) * G_ * H_);
    }

    // 5 FC heads summed, then log-softmax
    fc_kernel<<<cdiv(G_ * OUT_, B), B, 0, stream>>>(pooled, fc_w, fc_b, logits);
    log_softmax_kernel<<<cdiv(G_, B), B, 0, stream>>>(logits, out);
}